// OrthoLayer_55035710931331
// MI455X (gfx1250) — compile-verified
//
#include <hip/hip_runtime.h>

typedef __attribute__((ext_vector_type(2))) float v2f;
typedef __attribute__((ext_vector_type(8))) float v8f;

#define NF      256      // IN_FEATURES == OUT_FEATURES
#define NSTEPS  509      // Givens schedule length
#define NTHETA  32640    // (2L-1-S)*S/2 with L=256,S=255

// ---------- cross-lane helper (wave32, ds_bpermute) ----------
__device__ __forceinline__ float lane_read(float v, int srcLane) {
  int r = __builtin_amdgcn_ds_bpermute(srcLane << 2, __builtin_bit_cast(int, v));
  return __builtin_bit_cast(float, r);
}

// ---------- phase 0: cos/sin table ----------
__global__ void ortho_sincos(const float* __restrict__ thetas,
                             float2* __restrict__ cs) {
  int i = blockIdx.x * blockDim.x + threadIdx.x;
  if (i < NTHETA) {
    float s, c;
    sincosf(thetas[i], &s, &c);
    float2 r; r.x = c; r.y = s;
    cs[i] = r;
  }
}

// ---------- phase 1: build M^T, one wave per row of M ----------
// Row r of M = e_r^T * G_1 * ... * G_T, kept in 8 VGPRs per lane:
// lane l owns elements [8l, 8l+7]. Even steps: pairs (2m,2m+1) all
// in-lane. Odd steps: 3 in-lane pairs + 1 lane-boundary pair via
// ds_bpermute. Stored as Mt[c*NF + r] = M[r][c] (output-col major).
__global__ void ortho_build_m(const float2* __restrict__ cs,
                              float* __restrict__ Mt) {
  const int lane = threadIdx.x & 31;
  const int row  = blockIdx.x * (blockDim.x >> 5) + (threadIdx.x >> 5);

  float v[8];
#pragma unroll
  for (int j = 0; j < 8; ++j) v[j] = (8 * lane + j == row) ? 1.0f : 0.0f;

  int off = 0;
  for (int t = 0; t < NSTEPS; ++t) {
    const int startc = t & 1;
    const int endc   = (t < 255) ? (t + 2) : (510 - t);
    const int n      = (endc - startc) >> 1;

    if (startc == 0) {
      // pairs (2m, 2m+1): lane l handles m = 4l+q, q=0..3 -> (v[2q],v[2q+1])
#pragma unroll
      for (int q = 0; q < 4; ++q) {
        int m = 4 * lane + q;
        if (m < n) {
          float2 c = cs[off + m];
          float a = v[2 * q], b = v[2 * q + 1];
          v[2 * q]     = a * c.x - b * c.y;
          v[2 * q + 1] = a * c.y + b * c.x;
        }
      }
    } else {
      // pairs (2m+1, 2m+2): read boundary sources BEFORE any writes
      float bnext = lane_read(v[0], lane + 1);  // next lane's elem 8(l+1)
      float aprev = lane_read(v[7], lane - 1);  // prev lane's elem 8l-1
#pragma unroll
      for (int q = 0; q < 3; ++q) {
        int m = 4 * lane + q;                   // pair (8l+2q+1, 8l+2q+2)
        if (m < n) {
          float2 c = cs[off + m];
          float a = v[2 * q + 1], b = v[2 * q + 2];
          v[2 * q + 1] = a * c.x - b * c.y;
          v[2 * q + 2] = a * c.y + b * c.x;
        }
      }
      int mo = 4 * lane + 3;                    // pair (8l+7, 8l+8): our 'a'
      if (lane < 31 && mo < n) {
        float2 c = cs[off + mo];
        v[7] = v[7] * c.x - bnext * c.y;
      }
      int mi = 4 * lane - 1;                    // pair (8l-1, 8l): our 'b'
      if (lane > 0 && mi < n) {
        float2 c = cs[off + mi];
        v[0] = aprev * c.y + v[0] * c.x;
      }
    }
    off += n;
  }

#pragma unroll
  for (int j = 0; j < 8; ++j) Mt[(8 * lane + j) * NF + row] = v[j];
}

// ---------- phase 2: y = x * M + bias via V_WMMA_F32_16X16X4_F32 ----------
// One wave per 32x32 output tile: 2 A frags x 2 B frags -> 4 WMMAs per
// 4 b64 loads (1 load : 1 wmma), 64 K-steps, 256 chained WMMAs per wave.
// A layout (16x4 f32): lanes 0-15 -> M=lane, K={0,1}; lanes 16-31 -> K={2,3}.
// C/D layout: VGPR i -> row M = i (lanes 0-15) / i+8 (lanes 16-31), col = lane%16.
__global__ void ortho_wmma_gemm(const float* __restrict__ x,
                                const float* __restrict__ Mt,
                                const float* __restrict__ bias,
                                float* __restrict__ y, int nTiles) {
  const int lane = threadIdx.x & 31;
  const int wave = threadIdx.x >> 5;
  const int tile = blockIdx.x * (blockDim.x >> 5) + wave;
  if (tile >= nTiles) return;

  const int rowT = tile >> 3;         // 128 row tiles (4096/32)
  const int colT = tile & 7;          // 8 col tiles  (256/32)
  const int row0 = rowT * 32, col0 = colT * 32;

  const int half = lane >> 4;         // 0: K base 0, 1: K base 2
  const int lm   = lane & 15;
  const int kb   = half * 2;

  const float* arow0 = x  + (row0 + lm) * NF + kb;       // rows row0..+15
  const float* arow1 = arow0 + 16 * NF;                  // rows row0+16..+31
  const float* bcol0 = Mt + (col0 + lm) * NF + kb;       // cols col0..+15
  const float* bcol1 = bcol0 + 16 * NF;                  // cols col0+16..+31

  v8f acc00 = {0.f,0.f,0.f,0.f,0.f,0.f,0.f,0.f};
  v8f acc01 = acc00, acc10 = acc00, acc11 = acc00;

#pragma unroll 4
  for (int s = 0; s < NF / 4; ++s) {
    v2f a0 = *(const v2f*)(arow0 + 4 * s);
    v2f a1 = *(const v2f*)(arow1 + 4 * s);
    v2f b0 = *(const v2f*)(bcol0 + 4 * s);
    v2f b1 = *(const v2f*)(bcol1 + 4 * s);
    acc00 = __builtin_amdgcn_wmma_f32_16x16x4_f32(false, a0, false, b0,
                                                  (short)0, acc00, false, false);
    acc01 = __builtin_amdgcn_wmma_f32_16x16x4_f32(false, a0, false, b1,
                                                  (short)0, acc01, false, false);
    acc10 = __builtin_amdgcn_wmma_f32_16x16x4_f32(false, a1, false, b0,
                                                  (short)0, acc10, false, false);
    acc11 = __builtin_amdgcn_wmma_f32_16x16x4_f32(false, a1, false, b1,
                                                  (short)0, acc11, false, false);
  }

  const float bz0 = bias[col0 + lm];
  const float bz1 = bias[col0 + 16 + lm];
#pragma unroll
  for (int i = 0; i < 8; ++i) {
    const int m = i + half * 8;       // row within 16-row sub-tile
    float* yr0 = y + (row0 + m) * NF + col0 + lm;
    float* yr1 = y + (row0 + 16 + m) * NF + col0 + lm;
    yr0[0]  = acc00[i] + bz0;
    yr0[16] = acc01[i] + bz1;
    yr1[0]  = acc10[i] + bz0;
    yr1[16] = acc11[i] + bz1;
  }
}

extern "C" void kernel_launch(void* const* d_in, const int* in_sizes, int n_in,
                              void* d_out, int out_size, void* d_ws, size_t ws_size,
                              hipStream_t stream) {
  const float* x      = (const float*)d_in[0];
  const float* thetas = (const float*)d_in[1];
  const float* bias   = (const float*)d_in[2];
  float* out          = (float*)d_out;

  // workspace: [cos/sin table: 32640 float2][Mt: 256x256 float]
  float2* cs = (float2*)d_ws;
  size_t csBytes = (size_t)NTHETA * sizeof(float2);
  float* Mt = (float*)((char*)d_ws + ((csBytes + 255) & ~(size_t)255));

  ortho_sincos<<<(NTHETA + 255) / 256, 256, 0, stream>>>(thetas, cs);

  // 256 rows of M, 1 wave each -> 32 blocks x 8 waves
  ortho_build_m<<<NF / 8, 256, 0, stream>>>(cs, Mt);

  const int rows   = in_sizes[0] / NF;            // 4096
  const int nTiles = (rows / 32) * (NF / 32);     // 1024 tiles of 32x32
  const int wavesPerBlock = 8;                    // 256 threads
  ortho_wmma_gemm<<<(nTiles + wavesPerBlock - 1) / wavesPerBlock, 256, 0, stream>>>(
      x, Mt, bias, out, nTiles);
}